// simple_GCN_38757784879388
// MI455X (gfx1250) — compile-verified
//
#include <hip/hip_runtime.h>
#include <hip/hip_bf16.h>
#include <math.h>
#include <stdint.h>

// GCN on MI455X (gfx1250):
//   - 3 GEMMs via V_WMMA_F32_16X16X4_F32 with W staged in LDS
//     * 64-col W staged via GLOBAL_LOAD_ASYNC_TO_LDS_B128 (+ s_wait_asynccnt)
//     * K-loop is 2x ds_load + v_wmma, bank-conflict-free LDS stride
//   - degree + sym-norm scatter-add via global_atomic_add_f32 (unsafeAtomicAdd)
//   - global_prefetch_b8 lookahead on gathered feature rows
// Working set (~45MB) fits in 192MB L2 -> L2/atomic bound; fp32 WMMA keeps
// exact fp32 semantics while using the matrix pipe.

typedef float v2f __attribute__((ext_vector_type(2)));
typedef float v8f __attribute__((ext_vector_type(8)));

#define GEMM_WAVES 4

__global__ void k_fill(float* __restrict__ p, float v, int n) {
  int i = blockIdx.x * blockDim.x + threadIdx.x;
  if (i < n) p[i] = v;
}

// deg[dst] += 1 over real edges (deg pre-initialized to 1.0 for self-loops)
__global__ void k_deg(const int* __restrict__ dst, float* __restrict__ deg, int e) {
  int i = blockIdx.x * blockDim.x + threadIdx.x;
  if (i < e) unsafeAtomicAdd(&deg[dst[i]], 1.0f);
}

__global__ void k_rsqrt(float* __restrict__ d, int n) {
  int i = blockIdx.x * blockDim.x + threadIdx.x;
  if (i < n) {
    float v = d[i];
    d[i] = v > 0.0f ? rsqrtf(v) : 0.0f;
  }
}

// Out[nrows x ncols] = X[nrows x 64] @ W[64 x ncols] (+ bias per column).
// One wave per 16-row tile, 4 waves/block sharing W in LDS.
// nrows must be a multiple of 16 (N=50000 is). ncols <= 64.
__global__ void k_gemm64_wmma(const float* __restrict__ X, const float* __restrict__ W,
                              const float* __restrict__ bias, float* __restrict__ Out,
                              int nrows, int ncols) {
  // LDS: 64 rows x ldsStride floats, max 64*80*4 = 20 KB
  __shared__ float Wlds[64 * 80];

  const int ctiles = (ncols + 15) >> 4;
  const int pcols  = ctiles * 16;                    // padded column count
  // Half-waves read rows kbase and kbase+2: stride % 32 == 16 makes the two
  // 16-lane groups hit disjoint 16-bank sets (64 banks total).
  const int ldsStride = ((pcols & 31) == 16) ? pcols : pcols + 16;

  if (ncols == 64) {
    // Async global->LDS staging: 64 rows x 16 x 16B chunks = 1024 b128 copies.
    // VDST VGPR = LDS byte address, VADDR = 64-bit global address (GV mode).
    for (int idx = threadIdx.x; idx < 1024; idx += blockDim.x) {
      int r  = idx >> 4;
      int c4 = (idx & 15) * 4;
      const float* gptr = W + (size_t)r * 64 + c4;
      unsigned ldsoff = (unsigned)(uintptr_t)(&Wlds[r * 80 + c4]);
      asm volatile("global_load_async_to_lds_b128 %0, %1, off"
                   :
                   : "v"(ldsoff), "v"(gptr)
                   : "memory");
    }
    asm volatile("s_wait_asynccnt 0x0" ::: "memory");
  } else {
    // Scalar zero-padded staging (classifier: 64 x 40 -> 64 x 48).
    for (int idx = threadIdx.x; idx < 64 * pcols; idx += blockDim.x) {
      int r = idx / pcols;
      int c = idx - r * pcols;
      Wlds[r * ldsStride + c] = (c < ncols) ? W[(size_t)r * ncols + c] : 0.0f;
    }
  }
  __syncthreads();

  const int lane = threadIdx.x & 31;
  const int wave = threadIdx.x >> 5;
  const int tile = blockIdx.x * GEMM_WAVES + wave;
  const int row0 = tile * 16;
  if (row0 >= nrows) return;               // wave-uniform: EXEC all-1 for WMMA
  const int half = lane >> 4;              // 0: lanes 0-15, 1: lanes 16-31
  const int l16  = lane & 15;

  // Preload all 16 A fragments (A 16x4 tile per k-step: lane = M row,
  // half selects K in {0,1} vs {2,3} within each 4-wide K chunk).
  v2f afrag[16];
  const float* arow = X + (size_t)(row0 + l16) * 64 + half * 2;
#pragma unroll
  for (int kk = 0; kk < 16; ++kk) {
    afrag[kk].x = arow[kk * 4 + 0];
    afrag[kk].y = arow[kk * 4 + 1];
  }

  for (int ct = 0; ct < ctiles; ++ct) {
    const int col = ct * 16 + l16;
    float bv = 0.0f;
    if (bias != nullptr && col < ncols) bv = bias[col];
    v8f acc;
#pragma unroll
    for (int r = 0; r < 8; ++r) acc[r] = bv;

#pragma unroll
    for (int kk = 0; kk < 16; ++kk) {
      const int kbase = kk * 4 + half * 2;           // K pair for this half-wave
      // B 4x16 tile from LDS: lane = N (col), half selects K rows {0,1}/{2,3}
      v2f b;
      b.x = Wlds[kbase * ldsStride + col];
      b.y = Wlds[(kbase + 1) * ldsStride + col];
      acc = __builtin_amdgcn_wmma_f32_16x16x4_f32(
          /*neg_a=*/false, afrag[kk], /*neg_b=*/false, b,
          /*c_mod=*/(short)0, acc, /*reuse_a=*/false, /*reuse_b=*/false);
    }

    if (col < ncols) {
#pragma unroll
      for (int r = 0; r < 8; ++r) {
        const int row = row0 + half * 8 + r;         // C/D layout: VGPR r -> M=r / M=r+8
        Out[(size_t)row * ncols + col] = acc[r];
      }
    }
  }
}

// agg[i,:] = hw[i,:] * dinv[i]^2   (self-loop message; also initializes agg)
__global__ void k_agg_init(const float* __restrict__ hw, const float* __restrict__ dinv,
                           float* __restrict__ agg, int total) {
  int i = blockIdx.x * blockDim.x + threadIdx.x;
  if (i < total) {
    int row = i >> 6;
    float dv = dinv[row];
    agg[i] = hw[i] * dv * dv;
  }
}

// agg[dst,:] += hw[src,:] * dinv[src]*dinv[dst]  (one wave per edge, float2/lane)
__global__ void k_agg_edge(const float* __restrict__ hw, const float* __restrict__ dinv,
                           const int* __restrict__ src, const int* __restrict__ dst,
                           float* __restrict__ agg, int e) {
  long long gid = (long long)blockIdx.x * blockDim.x + threadIdx.x;
  int ed = (int)(gid >> 5);
  int c2 = ((int)gid & 31) * 2;
  if (ed < e) {
    int s = src[ed];
    int t = dst[ed];
    if (c2 == 0) {                      // lookahead prefetch of a future gathered row
      int la = ed + 16;
      if (la < e) __builtin_prefetch(&hw[(size_t)src[la] * 64], 0, 0);
    }
    float nm = dinv[s] * dinv[t];
    const float2* hrow = reinterpret_cast<const float2*>(hw + (size_t)s * 64);
    float2 v = hrow[c2 >> 1];
    float* arow = agg + (size_t)t * 64 + c2;
    unsafeAtomicAdd(arow + 0, v.x * nm);
    unsafeAtomicAdd(arow + 1, v.y * nm);
  }
}

// out[i] = tanh(agg[i] + bias[i % 64])
__global__ void k_bias_tanh(const float* __restrict__ agg, const float* __restrict__ bias,
                            float* __restrict__ out, int total) {
  int i = blockIdx.x * blockDim.x + threadIdx.x;
  if (i < total) {
    int d = i & 63;
    out[i] = tanhf(agg[i] + bias[d]);
  }
}

extern "C" void kernel_launch(void* const* d_in, const int* in_sizes, int n_in,
                              void* d_out, int out_size, void* d_ws, size_t ws_size,
                              hipStream_t stream) {
  const float* x  = (const float*)d_in[0];
  const int*   ei = (const int*)d_in[1];   // (2,E) int32: row0=src, row1=dst
  const float* W1 = (const float*)d_in[2];
  const float* b1 = (const float*)d_in[3];
  const float* W2 = (const float*)d_in[4];
  const float* b2 = (const float*)d_in[5];
  const float* Wc = (const float*)d_in[6];
  const float* bc = (const float*)d_in[7];

  const int n      = in_sizes[0] / 64;     // 50000
  const int e      = in_sizes[1] / 2;      // 800000
  const int nclass = in_sizes[6] / 64;     // 40
  const int* src = ei;
  const int* dst = ei + e;

  // workspace: dinv[n] | bufA[n*64] | bufB[n*64]  (~25.8 MB)
  float* dinv = (float*)d_ws;
  float* bufA = dinv + n;
  float* bufB = bufA + (size_t)n * 64;

  float* out_part = (float*)d_out;                       // [n, nclass]
  float* h_part   = out_part + (size_t)n * nclass;       // [n, 64]

  const int T   = 256;
  const int tot = n * 64;
  const long long ethreads = (long long)e * 32;          // one wave per edge

  // symmetric-normalization degrees (self-loop included via init to 1.0)
  k_fill <<<(n + T - 1) / T, T, 0, stream>>>(dinv, 1.0f, n);
  k_deg  <<<(e + T - 1) / T, T, 0, stream>>>(dst, dinv, e);
  k_rsqrt<<<(n + T - 1) / T, T, 0, stream>>>(dinv, n);

  const int tiles   = (n + 15) / 16;                     // 3125
  const int gblocks = (tiles + GEMM_WAVES - 1) / GEMM_WAVES;

  // ---- layer 1: h1 = tanh(Norm(x@W1) + b1)
  k_gemm64_wmma<<<gblocks, 32 * GEMM_WAVES, 0, stream>>>(x, W1, nullptr, bufB, n, 64);
  k_agg_init <<<(tot + T - 1) / T, T, 0, stream>>>(bufB, dinv, bufA, tot);
  k_agg_edge <<<(int)((ethreads + T - 1) / T), T, 0, stream>>>(bufB, dinv, src, dst, bufA, e);
  k_bias_tanh<<<(tot + T - 1) / T, T, 0, stream>>>(bufA, b1, bufB, tot);     // bufB = h1

  // ---- layer 2: h2 = tanh(Norm(h1@W2) + b2)  (h2 lives in d_out's second region)
  k_gemm64_wmma<<<gblocks, 32 * GEMM_WAVES, 0, stream>>>(bufB, W2, nullptr, bufA, n, 64);
  k_agg_init <<<(tot + T - 1) / T, T, 0, stream>>>(bufA, dinv, h_part, tot);
  k_agg_edge <<<(int)((ethreads + T - 1) / T), T, 0, stream>>>(bufA, dinv, src, dst, h_part, e);
  k_bias_tanh<<<(tot + T - 1) / T, T, 0, stream>>>(h_part, b2, h_part, tot);  // in-place

  // ---- classifier: out = h2@Wc + bc
  k_gemm64_wmma<<<gblocks, 32 * GEMM_WAVES, 0, stream>>>(h_part, Wc, bc, out_part, n, nclass);
}